// GCN_1726576853644
// MI455X (gfx1250) — compile-verified
//
#include <hip/hip_runtime.h>

typedef __attribute__((ext_vector_type(2))) float v2f;
typedef __attribute__((ext_vector_type(8))) float v8f;

#define D_FEAT 128
#define N_GRAPHS 2048
#define LDSA_STRIDE 132     // A tile row stride (dwords): conflict-free b64 frag reads
#define LDSB_PAIRSTRIDE 288 // W K-pair row stride (dwords): half-waves hit disjoint bank halves

// ---------------- degree / normalization ----------------
__global__ void k_deg_init(int* __restrict__ cnt, int n) {
  int i = blockIdx.x * 256 + threadIdx.x;
  if (i < n) cnt[i] = 1;                       // self loop
}
__global__ void k_deg_count(const int* __restrict__ dst, int* __restrict__ cnt, int E) {
  int e = blockIdx.x * 256 + threadIdx.x;
  if (e < E) atomicAdd(&cnt[dst[e]], 1);
}
__global__ void k_dinv(const int* __restrict__ cnt, float* __restrict__ dinv, int n) {
  int i = blockIdx.x * 256 + threadIdx.x;
  if (i < n) dinv[i] = rsqrtf((float)cnt[i]);  // cnt >= 1 always
}

// ---------------- CSR build: exclusive scan of in-degree ----------------
__global__ void k_scan1(const int* __restrict__ cnt, int* __restrict__ row_ptr,
                        int* __restrict__ partials, int n) {
  __shared__ int sh[256];
  int tx = threadIdx.x;
  int i = blockIdx.x * 256 + tx;
  int v = (i < n) ? (cnt[i] - 1) : 0;          // real-edge in-degree (minus self loop)
  sh[tx] = v;
  __syncthreads();
  for (int off = 1; off < 256; off <<= 1) {
    int t = (tx >= off) ? sh[tx - off] : 0;
    __syncthreads();
    sh[tx] += t;
    __syncthreads();
  }
  if (i < n) row_ptr[i] = sh[tx] - v;          // block-local exclusive
  if (tx == 255) partials[blockIdx.x] = sh[255];
}
__global__ void k_scan2(int* __restrict__ partials, int nb) {
  __shared__ int sh[512];
  int tx = threadIdx.x;
  int v = (tx < nb) ? partials[tx] : 0;
  sh[tx] = v;
  __syncthreads();
  for (int off = 1; off < 512; off <<= 1) {
    int t = (tx >= off) ? sh[tx - off] : 0;
    __syncthreads();
    sh[tx] += t;
    __syncthreads();
  }
  partials[tx] = sh[tx] - v;                   // exclusive block offsets
}
__global__ void k_scan3(int* __restrict__ row_ptr, const int* __restrict__ partials,
                        int* __restrict__ cursor, int n, int E) {
  int i = blockIdx.x * 256 + threadIdx.x;
  if (i < n) {
    int val = row_ptr[i] + partials[i >> 8];
    row_ptr[i] = val;
    cursor[i] = val;
  }
  if (i == 0) row_ptr[n] = E;
}
__global__ void k_fill(const int* __restrict__ src, const int* __restrict__ dst,
                       int* __restrict__ cursor, int* __restrict__ csr_src, int E) {
  int e = blockIdx.x * 256 + threadIdx.x;
  if (e < E) {
    int pos = atomicAdd(&cursor[dst[e]], 1);
    csr_src[pos] = src[e];
  }
}

// ---------------- GEMM: hW = hin @ W  (exact fp32 via V_WMMA_F32_16X16X4_F32) -------
// Block = 256 threads = 8 waves; each wave owns a 16-row M tile, all 128 N columns.
// A tile staged via GLOBAL_LOAD_ASYNC_TO_LDS_B128 (ASYNCcnt); W staged into a
// K-pair-interleaved layout so each B fragment is one aligned ds_load_b64.
__launch_bounds__(256)
__global__ void k_gemm(const float* __restrict__ hin, const float* __restrict__ W,
                       float* __restrict__ hW, int nrows) {
  __shared__ float w2[64 * LDSB_PAIRSTRIDE];       // 73728 B: [(k>>1)*288 + 2n + (k&1)]
  __shared__ float a_lds[8 * 16 * LDSA_STRIDE];    // 67584 B
  const int tid = threadIdx.x;
  const int wave = tid >> 5;
  const int lane = tid & 31;
  const int row0 = blockIdx.x * 128 + wave * 16;

  // 1) async-stage this wave's 16 A rows -> LDS (wave-private; own-wave wait suffices)
  float* amy = &a_lds[wave * 16 * LDSA_STRIDE];
  for (int i = lane; i < 512; i += 32) {           // 512 float4
    int r = i >> 5, c4 = i & 31;
    int gr = row0 + r;
    if (gr >= nrows) gr = nrows - 1;               // clamp tail; EXEC stays full
    const float* gsrc = hin + (size_t)gr * D_FEAT + c4 * 4;
    unsigned ldsa = (unsigned)(uintptr_t)(amy + r * LDSA_STRIDE + c4 * 4);
    asm volatile("global_load_async_to_lds_b128 %0, %1, off"
                 :: "v"(ldsa), "v"(gsrc) : "memory");
  }

  // 2) cooperative W staging, K-pair interleaved: w2[kp*288 + 2n + (k&1)]
  for (int p = tid; p < 4096; p += 256) {          // 64 kp x 64 n-pairs
    int kp = p >> 6;                               // K pair index
    int n0 = (p & 63) * 2;
    float2 r0 = *(const float2*)(W + (size_t)(2 * kp) * D_FEAT + n0);
    float2 r1 = *(const float2*)(W + (size_t)(2 * kp + 1) * D_FEAT + n0);
    float2 t0; t0.x = r0.x; t0.y = r1.x;
    float2 t1; t1.x = r0.y; t1.y = r1.y;
    *(float2*)&w2[kp * LDSB_PAIRSTRIDE + 2 * n0] = t0;
    *(float2*)&w2[kp * LDSB_PAIRSTRIDE + 2 * n0 + 2] = t1;
  }

  asm volatile("s_wait_asynccnt 0x0" ::: "memory");
  __syncthreads();

  v8f acc[8];
  const v8f zero = {0.f, 0.f, 0.f, 0.f, 0.f, 0.f, 0.f, 0.f};
#pragma unroll
  for (int j = 0; j < 8; j++) acc[j] = zero;

  const int mlane = lane & 15;
  const int koff = (lane >> 4) << 1;               // lanes 0-15 -> K{0,1}; 16-31 -> K{2,3}
  const float* arow = &amy[mlane * LDSA_STRIDE];

  for (int kk = 0; kk < 32; kk++) {
    const int kb = kk * 4 + koff;                  // even
    v2f a = *(const v2f*)(arow + kb);              // ds_load_b64, conflict-free
#pragma unroll
    for (int j = 0; j < 8; j++) {
      const int n = j * 16 + mlane;
      v2f b = *(const v2f*)&w2[(kb >> 1) * LDSB_PAIRSTRIDE + 2 * n];  // ds_load_b64
      acc[j] = __builtin_amdgcn_wmma_f32_16x16x4_f32(
          false, a, false, b, (short)0, acc[j], false, false);
    }
  }

  // D layout: VGPR v, lanes 0-15 -> row v, lanes 16-31 -> row v+8; N = lane&15
  const int rshift = (lane >> 4) * 8;
#pragma unroll
  for (int v = 0; v < 8; v++) {
    int r = row0 + v + rshift;
    if (r < nrows) {
#pragma unroll
      for (int j = 0; j < 8; j++)
        hW[(size_t)r * D_FEAT + j * 16 + mlane] = acc[j][v];
    }
  }
}

// ---------------- aggregation: gather over dst-CSR, fused self-loop + bias + relu ---
__global__ void k_aggregate(const float* __restrict__ hW, const float* __restrict__ dinv,
                            const int* __restrict__ row_ptr, const int* __restrict__ csr_src,
                            const float* __restrict__ bias, float* __restrict__ hout,
                            int nrows) {
  const int wave = threadIdx.x >> 5;
  const int lane = threadIdx.x & 31;
  const int n = blockIdx.x * 8 + wave;
  if (n >= nrows) return;

  const float dn = dinv[n];
  float4 sum = *(const float4*)(hW + (size_t)n * D_FEAT + lane * 4);
  const float selfw = dn * dn;
  sum.x *= selfw; sum.y *= selfw; sum.z *= selfw; sum.w *= selfw;

  const int beg = row_ptr[n];
  const int end = row_ptr[n + 1];
  for (int base = beg; base < end; base += 32) {
    int m = end - base;
    if (m > 32) m = 32;
    int s = 0;
    float ds = 0.f;
    if (lane < m) {
      s = csr_src[base + lane];
      ds = dinv[s];
    }
    for (int j = 0; j < m; j++) {                  // wave-uniform trip count
      int sj = __shfl(s, j, 32);
      float wj = __shfl(ds, j, 32) * dn;
      float4 v = *(const float4*)(hW + (size_t)sj * D_FEAT + lane * 4);
      sum.x += v.x * wj; sum.y += v.y * wj; sum.z += v.z * wj; sum.w += v.w * wj;
    }
  }
  const float4 b4 = *(const float4*)(bias + lane * 4);
  sum.x = fmaxf(sum.x + b4.x, 0.f);
  sum.y = fmaxf(sum.y + b4.y, 0.f);
  sum.z = fmaxf(sum.z + b4.z, 0.f);
  sum.w = fmaxf(sum.w + b4.w, 0.f);
  *(float4*)(hout + (size_t)n * D_FEAT + lane * 4) = sum;
}

// ---------------- global mean pool + final linear ----------------
__global__ void k_pool_clear(float* __restrict__ pool, float* __restrict__ pcnt) {
  int i = blockIdx.x * 256 + threadIdx.x;
  if (i < N_GRAPHS * D_FEAT) pool[i] = 0.f;
  if (i < N_GRAPHS) pcnt[i] = 0.f;
}
__global__ void k_pool_accum(const float* __restrict__ h, const int* __restrict__ batch,
                             float* __restrict__ pool, float* __restrict__ pcnt, int nrows) {
  const int wave = threadIdx.x >> 5;
  const int lane = threadIdx.x & 31;
  const int n = blockIdx.x * 8 + wave;
  if (n >= nrows) return;
  const int g = batch[n];
  const float4 v = *(const float4*)(h + (size_t)n * D_FEAT + lane * 4);
  float* p = pool + (size_t)g * D_FEAT + lane * 4;
  atomicAdd(p + 0, v.x);
  atomicAdd(p + 1, v.y);
  atomicAdd(p + 2, v.z);
  atomicAdd(p + 3, v.w);
  if (lane == 0) atomicAdd(&pcnt[g], 1.f);
}
__global__ void k_final(const float* __restrict__ pool, const float* __restrict__ pcnt,
                        const float* __restrict__ lin_w, const float* __restrict__ lin_b,
                        float* __restrict__ out) {
  const int wave = threadIdx.x >> 5;
  const int lane = threadIdx.x & 31;
  const int g = blockIdx.x * 8 + wave;
  if (g >= N_GRAPHS) return;
  const float inv = 1.f / fmaxf(pcnt[g], 1.f);
  const float4 p = *(const float4*)(pool + (size_t)g * D_FEAT + lane * 4);
  const float4 w = *(const float4*)(lin_w + lane * 4);
  float t = (p.x * w.x + p.y * w.y + p.z * w.z + p.w * w.w) * inv;
#pragma unroll
  for (int off = 16; off > 0; off >>= 1) t += __shfl_xor(t, off, 32);
  if (lane == 0) out[g] = t + lin_b[0];
}

// ---------------- driver ----------------
extern "C" void kernel_launch(void* const* d_in, const int* in_sizes, int n_in,
                              void* d_out, int out_size, void* d_ws, size_t ws_size,
                              hipStream_t stream) {
  const float* x     = (const float*)d_in[0];
  const int*   edge  = (const int*)d_in[1];
  const int*   batch = (const int*)d_in[2];
  const float* W     = (const float*)d_in[3];
  const float* bias  = (const float*)d_in[4];
  const float* lin_w = (const float*)d_in[5];
  const float* lin_b = (const float*)d_in[6];
  float* out = (float*)d_out;

  const int N = in_sizes[2];           // nodes
  const int E = in_sizes[1] / 2;       // edges
  const int* src = edge;
  const int* dst = edge + E;

  // carve workspace (256B aligned slabs)
  char* wsp = (char*)d_ws;
  auto alloc = [&](size_t bytes) -> void* {
    void* p = (void*)wsp;
    wsp += (bytes + 255) & ~(size_t)255;
    return p;
  };
  float* H        = (float*)alloc((size_t)N * D_FEAT * sizeof(float));
  float* hWbuf    = (float*)alloc((size_t)N * D_FEAT * sizeof(float));
  float* dinv     = (float*)alloc((size_t)N * sizeof(float));
  int*   cnt      = (int*)alloc((size_t)N * sizeof(int));
  int*   row_ptr  = (int*)alloc((size_t)(N + 1) * sizeof(int));
  int*   cursor   = (int*)alloc((size_t)N * sizeof(int));
  int*   csr_src  = (int*)alloc((size_t)E * sizeof(int));
  int*   partials = (int*)alloc(4096);
  float* pool     = (float*)alloc((size_t)N_GRAPHS * D_FEAT * sizeof(float));
  float* pcnt     = (float*)alloc((size_t)N_GRAPHS * sizeof(float));

  const int nbN = (N + 255) / 256;
  const int nbE = (E + 255) / 256;

  // normalization
  k_deg_init<<<nbN, 256, 0, stream>>>(cnt, N);
  k_deg_count<<<nbE, 256, 0, stream>>>(dst, cnt, E);
  k_dinv<<<nbN, 256, 0, stream>>>(cnt, dinv, N);

  // dst-CSR build
  k_scan1<<<nbN, 256, 0, stream>>>(cnt, row_ptr, partials, N);
  k_scan2<<<1, 512, 0, stream>>>(partials, nbN);
  k_scan3<<<nbN, 256, 0, stream>>>(row_ptr, partials, cursor, N, E);
  k_fill<<<nbE, 256, 0, stream>>>(src, dst, cursor, csr_src, E);

  // 3 GCN layers
  const float* hin = x;
  for (int l = 0; l < 3; l++) {
    k_gemm<<<(N + 127) / 128, 256, 0, stream>>>(hin, W + (size_t)l * D_FEAT * D_FEAT,
                                                hWbuf, N);
    k_aggregate<<<(N + 7) / 8, 256, 0, stream>>>(hWbuf, dinv, row_ptr, csr_src,
                                                 bias + (size_t)l * D_FEAT, H, N);
    hin = H;
  }

  // mean pool + linear head
  k_pool_clear<<<(N_GRAPHS * D_FEAT + 255) / 256, 256, 0, stream>>>(pool, pcnt);
  k_pool_accum<<<(N + 7) / 8, 256, 0, stream>>>(H, batch, pool, pcnt, N);
  k_final<<<N_GRAPHS / 8, 256, 0, stream>>>(pool, pcnt, lin_w, lin_b, out);
}